// MMDLoss_27307402068607
// MI455X (gfx1250) — compile-verified
//
#include <hip/hip_runtime.h>
#include <math.h>

// Problem constants (match the JAX reference)
#define N 4096
#define D 64
#define KK 5
#define TILES 256                       // N / 16
#define NTILES (TILES * TILES)          // 65536 16x16 tiles
#define WAVES_PER_BLOCK 8
#define BLOCK 256
#define GEMM_BLOCKS (NTILES / WAVES_PER_BLOCK)   // 8192
#define K_TARGET 8388607u               // (N*N - 1) / 2 : lower-median index

typedef __attribute__((ext_vector_type(2))) float v2f;
typedef __attribute__((ext_vector_type(8))) float v8f;

struct SelState {
  unsigned prefix;      // accumulated radix prefix of the median key
  unsigned kRemain;     // remaining rank within current prefix
  float    base_scale;  // -1 / median(xx)
  unsigned pad;
};

// Workspace layout (bytes)
#define WS_HIST  0                        // 2048 x u32
#define WS_STATE 8192                     // SelState
#define WS_PXX   16384                    // 8192 doubles
#define WS_PYY   (16384 + 1 * 65536)
#define WS_PZZ   (16384 + 2 * 65536)
#define WS_XN    (16384 + 3 * 65536)      // 4096 floats
#define WS_YN    (WS_XN + 16384)          // 4096 floats

// ---------------------------------------------------------------------------
// 16x16 squared-distance tile via V_WMMA_F32_16X16X4_F32 (one tile per wave).
// dist[m][n] = ||A_m||^2 + ||B_n||^2 - 2 * A_m . B_n
// A-matrix VGPR layout (16x4 f32): lanes 0-15 -> K = k0,k0+1 ; lanes 16-31 ->
// K = k0+2,k0+3 ; M = lane%16. B (4x16) mirrored over N. C/D: 8 VGPRs,
// element v: M = v + 8*(lane>=16), N = lane%16.
// ---------------------------------------------------------------------------
__device__ __forceinline__ v8f sqdist_tile(const float* __restrict__ A,
                                           const float* __restrict__ B,
                                           const float* __restrict__ an,
                                           const float* __restrict__ bn,
                                           int row0, int col0) {
  const int lane = threadIdx.x & 31;
  const int half = lane >> 4;   // 0 or 1
  const int l    = lane & 15;
  v8f acc = {0.f, 0.f, 0.f, 0.f, 0.f, 0.f, 0.f, 0.f};
  const float* arow = A + (size_t)(row0 + l) * D + half * 2;
  const float* brow = B + (size_t)(col0 + l) * D + half * 2;
#pragma unroll
  for (int k0 = 0; k0 < D; k0 += 4) {
    v2f a = *(const v2f*)(arow + k0);
    v2f b = *(const v2f*)(brow + k0);
    // D = A(16x4) * B(4x16) + C   -> v_wmma_f32_16x16x4_f32
    acc = __builtin_amdgcn_wmma_f32_16x16x4_f32(
        /*neg_a=*/false, a, /*neg_b=*/false, b,
        /*c_mod=*/(short)0, acc, /*reuse_a=*/false, /*reuse_b=*/false);
  }
  const float bnc = bn[col0 + l];
  v8f dst;
#pragma unroll
  for (int v = 0; v < 8; ++v) {
    dst[v] = an[row0 + (v + 8 * half)] + bnc - 2.0f * acc[v];
  }
  return dst;
}

// ---------------------------------------------------------------------------
// Init: zero histogram, reset selection state (run once per launch; ws is not
// re-poisoned between graph replays so we must re-init every call).
// ---------------------------------------------------------------------------
__global__ void init_kernel(unsigned* __restrict__ hist, SelState* __restrict__ st) {
  int idx = blockIdx.x * blockDim.x + threadIdx.x;
  if (idx < 2048) hist[idx] = 0u;
  if (idx == 0) {
    st->prefix = 0u;
    st->kRemain = K_TARGET;
    st->base_scale = 0.f;
    st->pad = 0u;
  }
}

// Row squared norms for x and y.
__global__ void norms_kernel(const float* __restrict__ x, const float* __restrict__ y,
                             float* __restrict__ xn, float* __restrict__ yn) {
  int i = blockIdx.x * blockDim.x + threadIdx.x;
  if (i >= N) return;
  float sx = 0.f, sy = 0.f;
#pragma unroll 4
  for (int k = 0; k < D; ++k) {
    float a = x[(size_t)i * D + k];
    float b = y[(size_t)i * D + k];
    sx += a * a;
    sy += b * b;
  }
  xn[i] = sx;
  yn[i] = sy;
}

// ---------------------------------------------------------------------------
// Radix-select histogram pass over xx (recomputed with WMMA each pass).
// Key = monotonic uint transform of the f32 distance.
// pass 0: digit = key[31:21]; pass 1: digit = key[20:10] under 11-bit prefix;
// pass 2: digit = key[9:0] under 22-bit prefix.
// ---------------------------------------------------------------------------
__global__ void hist_kernel(const float* __restrict__ X, const float* __restrict__ xn,
                            const SelState* __restrict__ st,
                            unsigned* __restrict__ hist, int pass) {
  __shared__ unsigned lh[2048];
  for (int i = threadIdx.x; i < 2048; i += BLOCK) lh[i] = 0u;
  __syncthreads();

  const int tile = blockIdx.x * WAVES_PER_BLOCK + (threadIdx.x >> 5);
  const int row0 = (tile >> 8) << 4;
  const int col0 = (tile & 255) << 4;
  v8f d = sqdist_tile(X, X, xn, xn, row0, col0);

  const unsigned prefix = st->prefix;
#pragma unroll
  for (int v = 0; v < 8; ++v) {
    unsigned b = __float_as_uint(d[v]);
    unsigned key = b ^ (((unsigned)((int)b >> 31)) | 0x80000000u);
    unsigned dig;
    bool ok;
    if (pass == 0)      { ok = true;                    dig = key >> 21; }
    else if (pass == 1) { ok = ((key >> 21) == prefix); dig = (key >> 10) & 0x7FFu; }
    else                { ok = ((key >> 10) == prefix); dig = key & 0x3FFu; }
    if (ok) atomicAdd(&lh[dig], 1u);
  }
  __syncthreads();
  for (int i = threadIdx.x; i < 2048; i += BLOCK) {
    unsigned c = lh[i];
    if (c) atomicAdd(&hist[i], c);
  }
}

// Single-block scan: locate the bin containing the k-th element, update
// prefix/kRemain, zero the histogram for the next pass. On the final pass the
// full 32-bit key is known -> exact lower median -> base_scale = -1/median.
__global__ void scan_kernel(SelState* __restrict__ st, unsigned* __restrict__ hist,
                            int nbins, int nbinsLog, int isFinal) {
  if (threadIdx.x == 0) {
    unsigned k = st->kRemain;
    unsigned cum = 0u;
    unsigned bin = (unsigned)(nbins - 1);
    for (int b = 0; b < nbins; ++b) {
      unsigned c = hist[b];
      if (k < cum + c) { bin = (unsigned)b; break; }
      cum += c;
    }
    st->kRemain = k - cum;
    unsigned prefix = (st->prefix << nbinsLog) | bin;
    st->prefix = prefix;
    if (isFinal) {
      unsigned key = prefix;
      unsigned bits = (key & 0x80000000u) ? (key ^ 0x80000000u) : ~key;
      float med = __uint_as_float(bits);
      st->base_scale = -1.0f / med;
    }
  }
  __syncthreads();
  for (int i = threadIdx.x; i < 2048; i += blockDim.x) hist[i] = 0u;
}

// ---------------------------------------------------------------------------
// Fused WMMA distance tile + multi-bandwidth exp sum + block reduction.
// maskDiag=1 zeroes diagonal contributions (kxx, kyy).
// ---------------------------------------------------------------------------
__global__ void kernelsum_kernel(const float* __restrict__ A, const float* __restrict__ B,
                                 const float* __restrict__ an, const float* __restrict__ bn,
                                 const float* __restrict__ kscales,
                                 const SelState* __restrict__ st,
                                 double* __restrict__ partials, int maskDiag) {
  __shared__ double red[BLOCK];
  const int tile = blockIdx.x * WAVES_PER_BLOCK + (threadIdx.x >> 5);
  const int row0 = (tile >> 8) << 4;
  const int col0 = (tile & 255) << 4;
  v8f d = sqdist_tile(A, B, an, bn, row0, col0);

  const float bs = st->base_scale;
  const float s0 = bs * kscales[0];
  const float s1 = bs * kscales[1];
  const float s2 = bs * kscales[2];
  const float s3 = bs * kscales[3];
  const float s4 = bs * kscales[4];

  const int lane = threadIdx.x & 31;
  const int half = lane >> 4;
  const int col = col0 + (lane & 15);
  double local = 0.0;
#pragma unroll
  for (int v = 0; v < 8; ++v) {
    int row = row0 + v + 8 * half;
    if (maskDiag && row == col) continue;
    float dv = d[v];
    float e = __expf(dv * s0) + __expf(dv * s1) + __expf(dv * s2) +
              __expf(dv * s3) + __expf(dv * s4);
    local += (double)e;
  }
  red[threadIdx.x] = local;
  __syncthreads();
  for (int off = BLOCK / 2; off > 0; off >>= 1) {
    if (threadIdx.x < off) red[threadIdx.x] += red[threadIdx.x + off];
    __syncthreads();
  }
  if (threadIdx.x == 0) partials[blockIdx.x] = red[0];
}

// Combine per-block partials into the scalar MMD loss.
__global__ void finalize_kernel(const double* __restrict__ pxx,
                                const double* __restrict__ pyy,
                                const double* __restrict__ pzz,
                                float* __restrict__ out) {
  __shared__ double rx[256], ry[256], rz[256];
  double sx = 0.0, sy = 0.0, sz = 0.0;
  for (int i = threadIdx.x; i < GEMM_BLOCKS; i += 256) {
    sx += pxx[i]; sy += pyy[i]; sz += pzz[i];
  }
  rx[threadIdx.x] = sx; ry[threadIdx.x] = sy; rz[threadIdx.x] = sz;
  __syncthreads();
  for (int off = 128; off > 0; off >>= 1) {
    if (threadIdx.x < off) {
      rx[threadIdx.x] += rx[threadIdx.x + off];
      ry[threadIdx.x] += ry[threadIdx.x + off];
      rz[threadIdx.x] += rz[threadIdx.x + off];
    }
    __syncthreads();
  }
  if (threadIdx.x == 0) {
    const double invK = 1.0 / (double)KK;
    const double offd = 1.0 / ((double)N * (double)(N - 1));
    double res = offd * (rx[0] * invK) + offd * (ry[0] * invK)
               - 2.0 * (rz[0] * invK) / ((double)N * (double)N);
    out[0] = (float)res;
  }
}

extern "C" void kernel_launch(void* const* d_in, const int* in_sizes, int n_in,
                              void* d_out, int out_size, void* d_ws, size_t ws_size,
                              hipStream_t stream) {
  (void)in_sizes; (void)n_in; (void)out_size; (void)ws_size;
  const float* x  = (const float*)d_in[0];
  const float* y  = (const float*)d_in[1];
  const float* ks = (const float*)d_in[2];
  float* out = (float*)d_out;

  char* ws = (char*)d_ws;
  unsigned* hist = (unsigned*)(ws + WS_HIST);
  SelState* st   = (SelState*)(ws + WS_STATE);
  double* pxx    = (double*)(ws + WS_PXX);
  double* pyy    = (double*)(ws + WS_PYY);
  double* pzz    = (double*)(ws + WS_PZZ);
  float* xn      = (float*)(ws + WS_XN);
  float* yn      = (float*)(ws + WS_YN);

  init_kernel<<<8, 256, 0, stream>>>(hist, st);
  norms_kernel<<<N / 256, 256, 0, stream>>>(x, y, xn, yn);

  // Exact lower-median of xx via 3-pass radix select (distances recomputed
  // with WMMA each pass; no N*N matrix is ever materialized).
  hist_kernel<<<GEMM_BLOCKS, BLOCK, 0, stream>>>(x, xn, st, hist, 0);
  scan_kernel<<<1, 256, 0, stream>>>(st, hist, 2048, 11, 0);
  hist_kernel<<<GEMM_BLOCKS, BLOCK, 0, stream>>>(x, xn, st, hist, 1);
  scan_kernel<<<1, 256, 0, stream>>>(st, hist, 2048, 11, 0);
  hist_kernel<<<GEMM_BLOCKS, BLOCK, 0, stream>>>(x, xn, st, hist, 2);
  scan_kernel<<<1, 256, 0, stream>>>(st, hist, 1024, 10, 1);

  // Fused distance + RBF kernel sums.
  kernelsum_kernel<<<GEMM_BLOCKS, BLOCK, 0, stream>>>(x, x, xn, xn, ks, st, pxx, 1);
  kernelsum_kernel<<<GEMM_BLOCKS, BLOCK, 0, stream>>>(y, y, yn, yn, ks, st, pyy, 1);
  kernelsum_kernel<<<GEMM_BLOCKS, BLOCK, 0, stream>>>(x, y, xn, yn, ks, st, pzz, 0);

  finalize_kernel<<<1, 256, 0, stream>>>(pxx, pyy, pzz, out);
}